// RelationalPathGNN_953482739768
// MI455X (gfx1250) — compile-verified
//
#include <hip/hip_runtime.h>
#include <math.h>

typedef __attribute__((ext_vector_type(2))) float v2f;
typedef __attribute__((ext_vector_type(8))) float v8f;

#define GN 20000
#define GE 160000
#define GEMB 64
#define GHID 128
#define GR 8

// ---------------- WMMA helper (f32 16x16x4, full precision) ----------------
__device__ inline v8f wmma4(v2f a, v2f b, v8f c) {
  // 8 args: (neg_a, A, neg_b, B, c_mod, C, reuse_a, reuse_b)
  return __builtin_amdgcn_wmma_f32_16x16x4_f32(false, a, false, b, (short)0, c,
                                               false, false);
}

__device__ inline float wred(float v) {
  v += __shfl_xor(v, 16, 32);
  v += __shfl_xor(v, 8, 32);
  v += __shfl_xor(v, 4, 32);
  v += __shfl_xor(v, 2, 32);
  v += __shfl_xor(v, 1, 32);
  return v;
}

__device__ inline void atomicMaxF(float* addr, float value) {
  if (value >= 0.0f)
    atomicMax((int*)addr, __float_as_int(value));
  else
    atomicMin((unsigned int*)addr, __float_as_uint(value));
}

// Async global -> LDS copy of 16 bytes (per lane). gfx1250 ASYNCcnt-tracked.
__device__ inline void async_g2lds_b128(unsigned lds_off, const float* gaddr) {
  asm volatile("global_load_async_to_lds_b128 %0, %1, off"
               :
               : "v"(lds_off), "v"(gaddr)
               : "memory");
}
__device__ inline void wait_asynccnt0() {
  asm volatile("s_wait_asynccnt 0x0" ::: "memory");
}

// ---------------- small utility kernels ----------------
__global__ void k_zero_u32(unsigned* p, int n) {
  int i = blockIdx.x * blockDim.x + threadIdx.x;
  if (i < n) p[i] = 0u;
}

__global__ void k_count_deg(const int* __restrict__ srcv, const int* __restrict__ dstv,
                            unsigned* __restrict__ oc, unsigned* __restrict__ ic, int n) {
  int e = blockIdx.x * blockDim.x + threadIdx.x;
  if (e >= n) return;
  atomicAdd(&oc[srcv[e]], 1u);
  atomicAdd(&ic[dstv[e]], 1u);
}

__global__ void k_scale(const unsigned* __restrict__ cnt, float* __restrict__ sc, int n) {
  int i = blockIdx.x * blockDim.x + threadIdx.x;
  if (i >= n) return;
  unsigned c = cnt[i];
  if (c < 1u) c = 1u;
  sc[i] = rsqrtf((float)c);
}

__global__ void k_count_rel(const int* __restrict__ eid, unsigned* __restrict__ rcount, int n) {
  int e = blockIdx.x * blockDim.x + threadIdx.x;
  if (e >= n) return;
  atomicAdd(&rcount[eid[e]], 1u);
}

__global__ void k_prefix8(const unsigned* __restrict__ rcount, unsigned* __restrict__ roff) {
  if (blockIdx.x == 0 && threadIdx.x == 0) {
    unsigned s = 0;
    for (int r = 0; r < GR; ++r) { roff[r] = s; s += rcount[r]; }
  }
}

__global__ void k_scatter_perm(const int* __restrict__ eid, const unsigned* __restrict__ roff,
                               unsigned* __restrict__ rcur, int* __restrict__ perm, int n) {
  int e = blockIdx.x * blockDim.x + threadIdx.x;
  if (e >= n) return;
  int r = eid[e];
  unsigned pos = roff[r] + atomicAdd(&rcur[r], 1u);
  perm[pos] = e;
}

// t[n] = dot(node_emb[n], Alo[0:64]) ; one wave per node
__global__ void k_node_dot(const float* __restrict__ ne, const float* __restrict__ Alo,
                           float* __restrict__ t, int n) {
  int wave = (blockIdx.x * blockDim.x + threadIdx.x) >> 5;
  int lane = threadIdx.x & 31;
  if (wave >= n) return;
  const float* p = ne + (size_t)wave * GEMB;
  float s = p[lane] * Alo[lane] + p[lane + 32] * Alo[lane + 32];
  s = wred(s);
  if (lane == 0) t[wave] = s;
}

__global__ void k_init_softmax(float* __restrict__ zmax, float* __restrict__ denom, int n) {
  int i = blockIdx.x * blockDim.x + threadIdx.x;
  if (i >= n) return;
  zmax[i] = -__builtin_inff();
  denom[i] = 0.0f;
}

// z[e] = leaky_relu(t[dst] + dot(m[e], Ahi)) ; atomic max into zmax[dst]
__global__ void k_attn_z(const float* __restrict__ m, const float* __restrict__ Ahi,
                         const float* __restrict__ t, const int* __restrict__ dstv,
                         float* __restrict__ z, float* __restrict__ zmax,
                         int nE, int outf) {
  int wave = (blockIdx.x * blockDim.x + threadIdx.x) >> 5;
  int lane = threadIdx.x & 31;
  if (wave >= nE) return;
  const float* me = m + (size_t)wave * outf;
  float s = 0.0f;
  for (int k = lane; k < outf; k += 32) s += me[k] * Ahi[k];
  s = wred(s);
  if (lane == 0) {
    float zz = t[dstv[wave]] + s;
    zz = (zz > 0.0f) ? zz : 0.01f * zz;  // leaky_relu, slope 0.01
    z[wave] = zz;
    atomicMaxF(&zmax[dstv[wave]], zz);
  }
}

// ez = exp(z - zmax[dst]) ; denom[dst] += ez ; overwrite z buffer with ez
__global__ void k_softmax_ez(float* __restrict__ zbuf, const float* __restrict__ zmax,
                             float* __restrict__ denom, const int* __restrict__ dstv, int nE) {
  int e = blockIdx.x * blockDim.x + threadIdx.x;
  if (e >= nE) return;
  int d = dstv[e];
  float ez = __expf(zbuf[e] - zmax[d]);
  zbuf[e] = ez;
  atomicAdd(&denom[d], ez);
}

// agg[dst] += (ez/denom[dst]) * m[e]  ; one wave per edge
__global__ void k_scatter_agg(const float* __restrict__ m, const float* __restrict__ ez,
                              const float* __restrict__ denom, const int* __restrict__ dstv,
                              float* __restrict__ agg, int nE, int outf) {
  int wave = (blockIdx.x * blockDim.x + threadIdx.x) >> 5;
  int lane = threadIdx.x & 31;
  if (wave >= nE) return;
  int d = dstv[wave];
  float coef = ez[wave] / denom[d];
  const float* me = m + (size_t)wave * outf;
  float* ag = agg + (size_t)d * outf;
  for (int k = lane; k < outf; k += 32) atomicAdd(&ag[k], coef * me[k]);
}

// out = relu(agg * in_scale + b)
__global__ void k_finalize(const float* __restrict__ agg, const float* __restrict__ in_scale,
                           const float* __restrict__ b, float* __restrict__ out,
                           int n, int outf) {
  int i = blockIdx.x * blockDim.x + threadIdx.x;
  if (i >= n * outf) return;
  int node = i / outf;
  int k = i - node * outf;
  float v = agg[i] * in_scale[node] + b[k];
  out[i] = (v > 0.0f) ? v : 0.0f;
}

// ---------------- dense self-loop GEMM: C[rows,outf] = A[rows,64] @ B[64,outf] ----------------
__global__ __launch_bounds__(32) void dense_gemm_wmma(const float* __restrict__ A,
                                                      const float* __restrict__ B,
                                                      float* __restrict__ C, int outf) {
  const int rt = blockIdx.x;      // row tile
  const int ct = blockIdx.y;      // col tile
  const int lane = threadIdx.x;
  const int hf = lane >> 4;       // lane-half selects K-half
  const int rc = lane & 15;       // row (A) / col (B, C)
  const float* Ab = A + (size_t)(rt * 16 + rc) * GEMB;
  const float* Bb = B + ct * 16 + rc;
  v8f c = {0.f, 0.f, 0.f, 0.f, 0.f, 0.f, 0.f, 0.f};
  for (int k = 0; k < GEMB; k += 4) {
    int kk = k + 2 * hf;
    v2f a, b;
    a[0] = Ab[kk];
    a[1] = Ab[kk + 1];
    b[0] = Bb[(size_t)kk * outf];
    b[1] = Bb[(size_t)(kk + 1) * outf];
    c = wmma4(a, b, c);
  }
#pragma unroll
  for (int v = 0; v < 8; ++v)
    C[(size_t)(rt * 16 + v + 8 * hf) * outf + ct * 16 + rc] = c[v];
}

// ---------------- grouped (relation-typed) GEMM over relation-sorted edge tiles ----------------
// x = [ h[src]*out_scale[src] (F_SRC) | edge_feat (64) | node_emb[dst] (64) ]  -> m = x @ W[eid]
// Staging goes global -> LDS via ASYNC loads (no VGPR round trip); the src-scale is a tiny
// LDS fix-up pass afterwards.
template <int IN_F, int F_SRC, int NW>
__global__ __launch_bounds__(NW * 32) void grouped_gemm_wmma(
    const float* __restrict__ h, const float* __restrict__ edge_feat,
    const float* __restrict__ node_emb, const float* __restrict__ W,
    const float* __restrict__ out_scale, const int* __restrict__ perm,
    const int* __restrict__ eid, const int* __restrict__ srcv,
    const int* __restrict__ dstv, float* __restrict__ m_out) {
  constexpr int OUT_F = NW * 16;
  constexpr int STR = IN_F + 4;  // +4 floats: bank-conflict-free; row pitch stays 16B-aligned
  __shared__ __align__(16) float xs[16 * STR];
  __shared__ int e_s[16];
  __shared__ int eid_s[16];
  __shared__ int src_s[16];
  __shared__ int dst_s[16];
  __shared__ float s_s[16];

  const int tid = threadIdx.x;
  const int tile = blockIdx.x;
  if (tid < 16) {
    int e = perm[tile * 16 + tid];
    e_s[tid] = e;
    eid_s[tid] = eid[e];
    int s = srcv[e];
    src_s[tid] = s;
    dst_s[tid] = dstv[e];
    s_s[tid] = out_scale[s];
  }
  __syncthreads();

  // Stage x tile straight into LDS with async b128 copies (16B chunks, all exactly aligned;
  // trip counts divide the block size so EXEC is all-ones on every issue).
  constexpr int CPR = IN_F / 4;
  const unsigned xs_base = (unsigned)(unsigned long long)(&xs[0]);  // LDS byte offset
  for (int idx = tid; idx < 16 * CPR; idx += NW * 32) {
    int row = idx / CPR;
    int kc = (idx - row * CPR) * 4;
    const float* gsrc;
    if (kc < F_SRC) {
      gsrc = h + (size_t)src_s[row] * F_SRC + kc;
    } else if (kc < F_SRC + GEMB) {
      gsrc = edge_feat + (size_t)e_s[row] * GEMB + (kc - F_SRC);
    } else {
      gsrc = node_emb + (size_t)dst_s[row] * GEMB + (kc - F_SRC - GEMB);
    }
    unsigned lds_off = xs_base + (unsigned)((row * STR + kc) * 4);
    async_g2lds_b128(lds_off, gsrc);
  }
  wait_asynccnt0();   // this wave's async copies are in LDS
  __syncthreads();    // ...and everyone else's

  // Fix-up: scale the h[src] region by out_deg^-0.5
  constexpr int SPR = F_SRC / 4;
  for (int idx = tid; idx < 16 * SPR; idx += NW * 32) {
    int row = idx / SPR;
    int kc = (idx - row * SPR) * 4;
    float sc = s_s[row];
    float4* p = (float4*)(xs + row * STR + kc);
    float4 v = *p;
    v.x *= sc; v.y *= sc; v.z *= sc; v.w *= sc;
    *p = v;
  }
  __syncthreads();

  const int wv = tid >> 5;   // wave -> 16-column tile
  const int lane = tid & 31;
  const int hf = lane >> 4;  // K-half
  const int rc = lane & 15;  // A row / B,C column
  const float* xrow = xs + rc * STR;

  // Edges are relation-sorted: tile spans contiguous non-decreasing eids.
  const int rlo = eid_s[0];
  const int rhi = eid_s[15];
  const float* Wc = W + (size_t)wv * 16 + rc;

  v8f c = {0.f, 0.f, 0.f, 0.f, 0.f, 0.f, 0.f, 0.f};
  if (rlo == rhi) {
    // Fast path: single relation in tile, no masking.
    const float* Wr = Wc + (size_t)rlo * IN_F * OUT_F;
#pragma unroll 4
    for (int k = 0; k < IN_F; k += 4) {
      int kk = k + 2 * hf;
      v2f a = *(const v2f*)(xrow + kk);  // ds_load_b64
      v2f b;
      b[0] = Wr[(size_t)kk * OUT_F];
      b[1] = Wr[(size_t)(kk + 1) * OUT_F];
      c = wmma4(a, b, c);
    }
  } else {
    // Boundary tile: loop relations, per-row mask on the A fragment.
    for (int r = rlo; r <= rhi; ++r) {
      float am = (eid_s[rc] == r) ? 1.0f : 0.0f;
      const float* Wr = Wc + (size_t)r * IN_F * OUT_F;
      for (int k = 0; k < IN_F; k += 4) {
        int kk = k + 2 * hf;
        v2f a = *(const v2f*)(xrow + kk);
        a[0] *= am;
        a[1] *= am;
        v2f b;
        b[0] = Wr[(size_t)kk * OUT_F];
        b[1] = Wr[(size_t)(kk + 1) * OUT_F];
        c = wmma4(a, b, c);
      }
    }
  }

#pragma unroll
  for (int v = 0; v < 8; ++v) {
    int rw = v + 8 * hf;
    int e = e_s[rw];
    m_out[(size_t)e * OUT_F + wv * 16 + rc] = c[v];
  }
}

// ---------------- host-side orchestration ----------------
extern "C" void kernel_launch(void* const* d_in, const int* in_sizes, int n_in,
                              void* d_out, int out_size, void* d_ws, size_t ws_size,
                              hipStream_t stream) {
  const float* node_emb  = (const float*)d_in[0];
  const float* edge_feat = (const float*)d_in[1];
  const float* W1 = (const float*)d_in[2];
  const float* A1 = (const float*)d_in[3];
  const float* b1 = (const float*)d_in[4];
  const float* L1 = (const float*)d_in[5];
  const float* W2 = (const float*)d_in[6];
  const float* A2 = (const float*)d_in[7];
  const float* b2 = (const float*)d_in[8];
  const float* L2 = (const float*)d_in[9];
  const int* eid  = (const int*)d_in[10];
  const int* srcv = (const int*)d_in[11];
  const int* dstv = (const int*)d_in[12];
  float* out = (float*)d_out;
  (void)in_sizes; (void)n_in; (void)out_size; (void)ws_size;

  char* wsp = (char*)d_ws;
  auto alloc = [&](size_t bytes) {
    char* p = wsp;
    wsp += (bytes + 255) & ~(size_t)255;
    return p;
  };
  int* perm        = (int*)alloc((size_t)GE * 4);
  unsigned* rmeta  = (unsigned*)alloc(96);  // rcount[8] | roff[8] | rcur[8]
  unsigned* rcount = rmeta;
  unsigned* roff   = rmeta + 8;
  unsigned* rcur   = rmeta + 16;
  unsigned* outcnt = (unsigned*)alloc((size_t)GN * 4);
  unsigned* incnt  = (unsigned*)alloc((size_t)GN * 4);
  float* out_scale = (float*)alloc((size_t)GN * 4);
  float* in_scale  = (float*)alloc((size_t)GN * 4);
  float* tbuf      = (float*)alloc((size_t)GN * 4);
  float* zmax      = (float*)alloc((size_t)GN * 4);
  float* denom     = (float*)alloc((size_t)GN * 4);
  float* zbuf      = (float*)alloc((size_t)GE * 4);
  float* mbuf      = (float*)alloc((size_t)GE * GHID * 4);
  float* agg       = (float*)alloc((size_t)GN * GHID * 4);
  float* h1        = (float*)alloc((size_t)GN * GHID * 4);

  const int TB = 256;
  const int gN  = (GN + TB - 1) / TB;
  const int gE  = (GE + TB - 1) / TB;
  const int gNw = (GN * 32 + TB - 1) / TB;   // wave-per-node
  const int gEw = (GE * 32 + TB - 1) / TB;   // wave-per-edge

  // ---- one-time prep: degrees, scales, relation sort ----
  k_zero_u32<<<gN, TB, 0, stream>>>(outcnt, GN);
  k_zero_u32<<<gN, TB, 0, stream>>>(incnt, GN);
  k_zero_u32<<<1, 32, 0, stream>>>(rmeta, 24);
  k_count_deg<<<gE, TB, 0, stream>>>(srcv, dstv, outcnt, incnt, GE);
  k_scale<<<gN, TB, 0, stream>>>(outcnt, out_scale, GN);
  k_scale<<<gN, TB, 0, stream>>>(incnt, in_scale, GN);
  k_count_rel<<<gE, TB, 0, stream>>>(eid, rcount, GE);
  k_prefix8<<<1, 1, 0, stream>>>(rcount, roff);
  k_scatter_perm<<<gE, TB, 0, stream>>>(eid, roff, rcur, perm, GE);

  // ---- layer 1: h = node_emb (F_SRC=64), W1[8,192,128] ----
  k_node_dot<<<gNw, TB, 0, stream>>>(node_emb, A1, tbuf, GN);
  dense_gemm_wmma<<<dim3(GN / 16, GHID / 16), 32, 0, stream>>>(node_emb, L1, agg, GHID);
  grouped_gemm_wmma<192, 64, 8><<<GE / 16, 256, 0, stream>>>(
      node_emb, edge_feat, node_emb, W1, out_scale, perm, eid, srcv, dstv, mbuf);
  k_init_softmax<<<gN, TB, 0, stream>>>(zmax, denom, GN);
  k_attn_z<<<gEw, TB, 0, stream>>>(mbuf, A1 + GEMB, tbuf, dstv, zbuf, zmax, GE, GHID);
  k_softmax_ez<<<gE, TB, 0, stream>>>(zbuf, zmax, denom, dstv, GE);
  k_scatter_agg<<<gEw, TB, 0, stream>>>(mbuf, zbuf, denom, dstv, agg, GE, GHID);
  k_finalize<<<(GN * GHID + TB - 1) / TB, TB, 0, stream>>>(agg, in_scale, b1, h1, GN, GHID);

  // ---- layer 2: h = h1 (F_SRC=128), W2[8,256,64] ----
  k_node_dot<<<gNw, TB, 0, stream>>>(node_emb, A2, tbuf, GN);
  dense_gemm_wmma<<<dim3(GN / 16, GEMB / 16), 32, 0, stream>>>(node_emb, L2, agg, GEMB);
  grouped_gemm_wmma<256, 128, 4><<<GE / 16, 128, 0, stream>>>(
      h1, edge_feat, node_emb, W2, out_scale, perm, eid, srcv, dstv, mbuf);
  k_init_softmax<<<gN, TB, 0, stream>>>(zmax, denom, GN);
  k_attn_z<<<gEw, TB, 0, stream>>>(mbuf, A2 + GEMB, tbuf, dstv, zbuf, zmax, GE, GEMB);
  k_softmax_ez<<<gE, TB, 0, stream>>>(zbuf, zmax, denom, dstv, GE);
  k_scatter_agg<<<gEw, TB, 0, stream>>>(mbuf, zbuf, denom, dstv, agg, GE, GEMB);
  k_finalize<<<(GN * GEMB + TB - 1) / TB, TB, 0, stream>>>(agg, in_scale, b2, out, GN, GEMB);
}